// GCNANE_58789512348191
// MI455X (gfx1250) — compile-verified
//
#include <hip/hip_runtime.h>

typedef __attribute__((ext_vector_type(2))) float v2f;
typedef __attribute__((ext_vector_type(8))) float v8f;

#define N_NODE  100000
#define N_ATTRI 10000
#define NTOT    110000
#define NEDGE   2000000
#define NFEAT   128
#define NHID    128
#define NHID2   64

// ---------------------------------------------------------------------------
// GEMM: out[M x NOUT] = concat(A0,A1)[M x 128] @ W[128 x NOUT]
// One block = 16 rows of M (A tile staged in LDS, shared by all waves).
// One wave = one 16x16 output tile, K-loop of 32x v_wmma_f32_16x16x4_f32.
// M = 110000 = 6875 * 16 exactly, so no tail handling needed.
// ---------------------------------------------------------------------------
template<int NOUT>
__global__ __launch_bounds__(32 * (NOUT / 16))
void gcn_gemm_wmma(const float* __restrict__ A0, const float* __restrict__ A1,
                   const float* __restrict__ W, float* __restrict__ out,
                   int nsplit)
{
    __shared__ float Alds[16 * NFEAT];            // 8 KB A tile
    constexpr int NT = 32 * (NOUT / 16);
    const int tid = threadIdx.x;
    const int m0  = blockIdx.x * 16;

    // Cooperative, coalesced load of the 16x128 A tile (handles the concat).
    for (int i = tid; i < 16 * NFEAT; i += NT) {
        const int r = m0 + (i >> 7);
        const int c = i & (NFEAT - 1);
        Alds[i] = (r < nsplit) ? A0[(size_t)r * NFEAT + c]
                               : A1[(size_t)(r - nsplit) * NFEAT + c];
    }
    __syncthreads();

    const int lane = tid & 31;
    const int n0   = (tid >> 5) * 16;             // wave -> 16-column tile
    const int l15  = lane & 15;                   // M index (A) / N index (B)
    const int hlf  = lane >> 4;                   // half-wave select
    const int koff = hlf * 2;                     // ISA 32-bit A layout: K=0,1 | K=2,3

    v8f acc = {};
    #pragma unroll 4
    for (int k = 0; k < NFEAT; k += 4) {
        v2f a, b;
        a.x = Alds[l15 * NFEAT + k + koff];
        a.y = Alds[l15 * NFEAT + k + koff + 1];
        b.x = W[(size_t)(k + koff)     * NOUT + n0 + l15];
        b.y = W[(size_t)(k + koff + 1) * NOUT + n0 + l15];
        // 8 args: (neg_a, A, neg_b, B, c_mod, C, reuse_a, reuse_b)
        acc = __builtin_amdgcn_wmma_f32_16x16x4_f32(
            false, a, false, b, (short)0, acc, false, false);
    }

    // C/D layout: VGPR i holds row M = i + 8*hlf, col N = l15.
    #pragma unroll
    for (int i = 0; i < 8; ++i)
        out[(size_t)(m0 + i + 8 * hlf) * NOUT + n0 + l15] = acc[i];
}

// ---------------------------------------------------------------------------
// Edge-parallel SpMM with atomics: agg[row] += val * sup[col].
// One wave per edge: coalesced 128B gathers from L2-resident support matrix,
// coalesced global_atomic_add_f32 scatter.
// ---------------------------------------------------------------------------
template<int NOUT>
__global__ __launch_bounds__(256)
void gcn_spmm_atomic(const int* __restrict__ rows, const int* __restrict__ cols,
                     const float* __restrict__ vals,
                     const float* __restrict__ sup, float* __restrict__ agg)
{
    const long flat = (long)blockIdx.x * blockDim.x + threadIdx.x;
    const int  e    = (int)(flat >> 5);
    const int  lane = threadIdx.x & 31;
    if (e >= NEDGE) return;

    const int   r = rows[e];
    const int   c = cols[e];
    const float v = vals[e];
    const float* __restrict__ s = sup + (size_t)c * NOUT;
    float* o = agg + (size_t)r * NOUT;

    #pragma unroll
    for (int j = 0; j < NOUT / 32; ++j)
        atomicAdd(o + lane + 32 * j, v * s[lane + 32 * j]);
}

__global__ __launch_bounds__(256)
void gcn_zero(float* __restrict__ p, long n)
{
    const long i = (long)blockIdx.x * blockDim.x + threadIdx.x;
    if (i < n) p[i] = 0.0f;
}

template<int NOUT>
__global__ __launch_bounds__(256)
void gcn_bias_relu(const float* __restrict__ x, const float* __restrict__ b,
                   float* __restrict__ y, long n)
{
    const long i = (long)blockIdx.x * blockDim.x + threadIdx.x;
    if (i < n) {
        const float t = x[i] + b[i & (NOUT - 1)];
        y[i] = t > 0.0f ? t : 0.0f;
    }
}

// ---------------------------------------------------------------------------
extern "C" void kernel_launch(void* const* d_in, const int* in_sizes, int n_in,
                              void* d_out, int out_size, void* d_ws, size_t ws_size,
                              hipStream_t stream)
{
    const int*   edge_index  = (const int*)  d_in[0];   // [2, E] int32
    const float* edge_values = (const float*)d_in[1];   // [E]
    const float* emb_node    = (const float*)d_in[2];   // [100000, 128]
    const float* emb_attri   = (const float*)d_in[3];   // [10000, 128]
    const float* W1          = (const float*)d_in[4];   // [128, 128]
    const float* b1          = (const float*)d_in[5];   // [128]
    const float* W2          = (const float*)d_in[6];   // [128, 64]
    const float* b2          = (const float*)d_in[7];   // [64]
    float*       out         = (float*)d_out;           // [110000, 64]

    const int* rows = edge_index;           // edge_index[0]
    const int* cols = edge_index + NEDGE;   // edge_index[1]

    float* bufA = (float*)d_ws;                          // support1 / support2
    float* bufB = bufA + (size_t)NTOT * NHID;            // h1 / agg2

    const long n1 = (long)NTOT * NHID;   // 14,080,000
    const long n2 = (long)NTOT * NHID2;  //  7,040,000
    const int  mblocks    = NTOT / 16;                   // 6875
    const int  spmmBlocks = (int)(((long)NEDGE * 32 + 255) / 256); // 250000

    // ---- Layer 1 ----
    gcn_gemm_wmma<NHID><<<mblocks, 256, 0, stream>>>(emb_node, emb_attri, W1,
                                                     bufA, N_NODE);
    gcn_zero<<<(int)((n1 + 255) / 256), 256, 0, stream>>>(bufB, n1);
    gcn_spmm_atomic<NHID><<<spmmBlocks, 256, 0, stream>>>(rows, cols, edge_values,
                                                          bufA, bufB);
    gcn_bias_relu<NHID><<<(int)((n1 + 255) / 256), 256, 0, stream>>>(bufB, b1,
                                                                     bufB, n1);

    // ---- Layer 2 ----
    gcn_gemm_wmma<NHID2><<<mblocks, 128, 0, stream>>>(bufB, bufB, W2,
                                                      bufA, NTOT);
    gcn_zero<<<(int)((n2 + 255) / 256), 256, 0, stream>>>(bufB, n2);
    gcn_spmm_atomic<NHID2><<<spmmBlocks, 256, 0, stream>>>(rows, cols, edge_values,
                                                           bufA, bufB);
    gcn_bias_relu<NHID2><<<(int)((n2 + 255) / 256), 256, 0, stream>>>(bufB, b2,
                                                                      out, n2);
}